// PainnMessage_37211596652607
// MI455X (gfx1250) — compile-verified
//
#include <hip/hip_runtime.h>
#include <math.h>

#define NN   50000
#define EE   800000
#define S    64
#define S3   192
#define NRAD 20
#define CUTR 5.0f
#define PI_F 3.14159265358979323846f

#define WPAD 66   // padded k-stride for transposed 64-deep weights (bank-conflict-free)
#define FPAD 22   // padded k-stride for transposed 20-deep Wf

typedef __attribute__((ext_vector_type(2))) float v2f;
typedef __attribute__((ext_vector_type(8))) float v8f;

__device__ __forceinline__ v8f wmma_f32(v2f a, v2f b, v8f c) {
    // D = A(16x4 f32) * B(4x16 f32) + C(16x16 f32), full f32 precision
    return __builtin_amdgcn_wmma_f32_16x16x4_f32(false, a, false, b, (short)0, c, false, false);
}

__device__ __forceinline__ v8f splat8(float x) {
    v8f v = {x, x, x, x, x, x, x, x};
    return v;
}

__device__ __forceinline__ float fast_rcp(float x) {
    return __builtin_amdgcn_rcpf(x);   // single v_rcp_f32 (1 ulp) instead of div sequence
}

// ---------------------------------------------------------------------------
// Kernel 0: out = concat(node_scalar, node_vector); residuals accumulate on top
// ---------------------------------------------------------------------------
__global__ __launch_bounds__(256) void init_out(const float* __restrict__ ns,
                                                const float* __restrict__ nv,
                                                float* __restrict__ out) {
    size_t i   = (size_t)blockIdx.x * blockDim.x + threadIdx.x;
    size_t nS4 = (size_t)NN * S / 4;
    size_t nV4 = (size_t)NN * 3 * S / 4;
    if (i < nS4) {
        ((float4*)out)[i] = ((const float4*)ns)[i];
    } else if (i < nS4 + nV4) {
        ((float4*)out)[i] = ((const float4*)nv)[i - nS4];
    }
}

// ---------------------------------------------------------------------------
// Kernel 1: scalar_out = silu(node_scalar @ W1 + b1) @ W2 + b2   (N x 192)
// 8 waves/block, each wave owns a 16-row tile. WMMA f32 16x16x4.
// Weights live transposed+padded in LDS so each B fragment is one ds_load_b64
// (merged by the compiler into ds_load_2addr_b64), and all 16 k-step fragments
// are prefetched before the back-to-back WMMA chain.
// ---------------------------------------------------------------------------
__global__ __launch_bounds__(256) void node_mlp(const float* __restrict__ ns,
                                                const float* __restrict__ W1,
                                                const float* __restrict__ b1,
                                                const float* __restrict__ W2,
                                                const float* __restrict__ b2,
                                                float* __restrict__ so) {
    __shared__ __align__(16) float sW1t[64 * WPAD];    // W1^T  [n][k]  16.9 KB
    __shared__ __align__(16) float sW2t[192 * WPAD];   // W2^T  [n][k]  50.7 KB
    __shared__ float sB1[64];
    __shared__ float sB2[192];
    __shared__ __align__(16) float sH[8][16 * WPAD];   // per-wave H stage, 33.8 KB

    const int tid = threadIdx.x;
    for (int i = tid; i < 64 * 64; i += 256) {
        const int k = i >> 6, n = i & 63;
        sW1t[n * WPAD + k] = W1[i];
    }
    for (int i = tid; i < 64 * 192; i += 256) {
        const int k = i / 192, n = i - k * 192;
        sW2t[n * WPAD + k] = W2[i];
    }
    if (tid < 64)  sB1[tid] = b1[tid];
    if (tid < 192) sB2[tid] = b2[tid];

    const int wave = tid >> 5;
    const int lane = tid & 31;
    const int lm   = lane & 15;   // A: row M    B/C: col N
    const int hi   = lane >> 4;   // K-half select / C row-half
    const int tile = blockIdx.x * 8 + wave;
    const int row0 = tile * 16;

    // A-matrix registers for GEMM1 (batched global b64 loads)
    int rload = row0 + lm;
    if (rload > NN - 1) rload = NN - 1;
    const float* arow = ns + (size_t)rload * S;
    v2f areg[16];
#pragma unroll
    for (int kk = 0; kk < 16; ++kk)
        areg[kk] = *(const v2f*)(arow + 4 * kk + 2 * hi);

    __syncthreads();  // weights staged

    float* Hs = &sH[wave][0];
#pragma unroll
    for (int t = 0; t < 4; ++t) {
        const int col = t * 16 + lm;
        // prefetch all B fragments (ds_load_2addr_b64 x8, one wait)
        v2f breg[16];
#pragma unroll
        for (int kk = 0; kk < 16; ++kk)
            breg[kk] = *(const v2f*)(sW1t + col * WPAD + 4 * kk + 2 * hi);
        v8f c = splat8(sB1[col]);
#pragma unroll
        for (int kk = 0; kk < 16; ++kk)
            c = wmma_f32(areg[kk], breg[kk], c);
        // SiLU (x * rcp(1+exp(-x)), one v_exp + one v_rcp), stage H into LDS
#pragma unroll
        for (int j = 0; j < 8; ++j) {
            float x = c[j];
            float h = x * fast_rcp(1.0f + __expf(-x));
            Hs[(j + 8 * hi) * WPAD + col] = h;
        }
    }
    __syncthreads();  // H staged

    v2f aH[16];
#pragma unroll
    for (int kk = 0; kk < 16; ++kk)
        aH[kk] = *(const v2f*)(Hs + lm * WPAD + 4 * kk + 2 * hi);

#pragma unroll
    for (int t = 0; t < 12; ++t) {
        const int col = t * 16 + lm;
        v2f breg[16];
#pragma unroll
        for (int kk = 0; kk < 16; ++kk)
            breg[kk] = *(const v2f*)(sW2t + col * WPAD + 4 * kk + 2 * hi);
        v8f c = splat8(sB2[col]);
#pragma unroll
        for (int kk = 0; kk < 16; ++kk)
            c = wmma_f32(aH[kk], breg[kk], c);
#pragma unroll
        for (int j = 0; j < 8; ++j) {
            const int r = row0 + j + 8 * hi;
            if (r < NN) so[(size_t)r * S3 + col] = c[j];
        }
    }
}

// ---------------------------------------------------------------------------
// Kernel 2: per-edge RBF filter GEMM (K=20, WMMA f32) fused with gather of
// scalar_out/node_vector and atomic scatter of messages into d_out.
// ---------------------------------------------------------------------------
struct EdgeP {
    float cut, d, ux, uy, uz;
    int   src, dst, pad;
};

__global__ __launch_bounds__(256) void edge_msg(const float* __restrict__ nv,
                                                const long long* __restrict__ edge,
                                                const float* __restrict__ ediff,
                                                const float* __restrict__ edist,
                                                const float* __restrict__ Wf,
                                                const float* __restrict__ bf,
                                                const float* __restrict__ so,
                                                float* __restrict__ outS,
                                                float* __restrict__ outV) {
    __shared__ __align__(16) float sWft[S3 * FPAD];  // Wf^T [n][k], 16.9 KB
    __shared__ float sBf[S3];
    __shared__ EdgeP sP[8][16];                      // 4 KB per-wave edge params

    const int tid = threadIdx.x;
    for (int i = tid; i < NRAD * S3; i += 256) {
        const int k = i / S3, n = i - k * S3;
        sWft[n * FPAD + k] = Wf[i];
    }
    if (tid < S3) sBf[tid] = bf[tid];

    const int wave = tid >> 5;
    const int lane = tid & 31;
    const int lm   = lane & 15;
    const int hi   = lane >> 4;
    const long long tile = (long long)blockIdx.x * 8 + wave;

    if (lane < 16) {
        const long long e  = tile * 16 + lane;
        const bool valid   = (e < (long long)EE);
        const long long el = valid ? e : (long long)(EE - 1);
        const float d   = edist[el];
        const float inv = fast_rcp(d);
        EdgeP p;
        p.d   = d;
        p.cut = (valid && (d < CUTR)) ? (0.5f * (__cosf(d * (PI_F / CUTR)) + 1.0f)) : 0.0f;
        p.ux  = ediff[el * 3 + 0] * inv;
        p.uy  = ediff[el * 3 + 1] * inv;
        p.uz  = ediff[el * 3 + 2] * inv;
        p.src = (int)edge[el * 2 + 1];
        p.dst = valid ? (int)edge[el * 2 + 0] : -1;
        p.pad = 0;
        sP[wave][lane] = p;
    }
    __syncthreads();

    // RBF A-matrix in registers: rbf[e,k] = sin(d*(k+1)*pi/5)/d, K=20 = 5 steps exactly
    const float dr   = sP[wave][lm].d;
    const float invd = fast_rcp(dr);
    v2f a[5];
#pragma unroll
    for (int kk = 0; kk < 5; ++kk) {
        const int klo = 4 * kk + 2 * hi;
        a[kk].x = __sinf(dr * (float)(klo + 1) * (PI_F / CUTR)) * invd;
        a[kk].y = __sinf(dr * (float)(klo + 2) * (PI_F / CUTR)) * invd;
    }

    // 4 column groups of 16; gate_state / gate_edge / message_scalar share columns
#pragma unroll
    for (int g = 0; g < 4; ++g) {
        const int col = g * 16 + lm;
        // prefetch all 15 B fragments (ds_load_b64 each), then the WMMA chain
        v2f bS[5], bE[5], bM[5];
#pragma unroll
        for (int kk = 0; kk < 5; ++kk) {
            const int klo = 4 * kk + 2 * hi;
            bS[kk] = *(const v2f*)(sWft + (col)       * FPAD + klo);
            bE[kk] = *(const v2f*)(sWft + (64 + col)  * FPAD + klo);
            bM[kk] = *(const v2f*)(sWft + (128 + col) * FPAD + klo);
        }
        v8f cS = splat8(sBf[col]);        // gate_state  (cols 0..63)
        v8f cE = splat8(sBf[64 + col]);   // gate_edge   (cols 64..127)
        v8f cM = splat8(sBf[128 + col]);  // msg_scalar  (cols 128..191)
#pragma unroll
        for (int kk = 0; kk < 5; ++kk) {
            cS = wmma_f32(a[kk], bS[kk], cS);
            cE = wmma_f32(a[kk], bE[kk], cE);
            cM = wmma_f32(a[kk], bM[kk], cM);
        }
#pragma unroll
        for (int j = 0; j < 8; ++j) {
            const int r = j + 8 * hi;          // edge row within tile
            EdgeP p = sP[wave][r];
            if (p.dst >= 0) {
                const float* soRow = so + (size_t)p.src * S3;
                const float vS = cS[j] * p.cut * soRow[col];
                const float vE = cE[j] * p.cut * soRow[64 + col];
                const float vM = cM[j] * p.cut * soRow[128 + col];
                const float* nvRow = nv + (size_t)p.src * 3 * S;
                atomicAdd(outS + (size_t)p.dst * S + col, vM);
                atomicAdd(outV + ((size_t)p.dst * 3 + 0) * S + col, fmaf(nvRow[col],         vS, vE * p.ux));
                atomicAdd(outV + ((size_t)p.dst * 3 + 1) * S + col, fmaf(nvRow[S + col],     vS, vE * p.uy));
                atomicAdd(outV + ((size_t)p.dst * 3 + 2) * S + col, fmaf(nvRow[2 * S + col], vS, vE * p.uz));
            }
        }
    }
}

// ---------------------------------------------------------------------------
extern "C" void kernel_launch(void* const* d_in, const int* in_sizes, int n_in,
                              void* d_out, int out_size, void* d_ws, size_t ws_size,
                              hipStream_t stream) {
    const float*     ns    = (const float*)d_in[0];
    const float*     nv    = (const float*)d_in[1];
    const long long* edge  = (const long long*)d_in[2];
    const float*     ediff = (const float*)d_in[3];
    const float*     edist = (const float*)d_in[4];
    const float*     W1    = (const float*)d_in[5];
    const float*     b1    = (const float*)d_in[6];
    const float*     W2    = (const float*)d_in[7];
    const float*     b2    = (const float*)d_in[8];
    const float*     Wf    = (const float*)d_in[9];
    const float*     bf    = (const float*)d_in[10];

    float* out  = (float*)d_out;
    float* outS = out;                         // N x 64
    float* outV = out + (size_t)NN * S;        // N x 3 x 64
    float* so   = (float*)d_ws;                // scalar_out: N x 192 f32 (38.4 MB)

    const size_t tot4 = ((size_t)NN * S + (size_t)NN * 3 * S) / 4;
    init_out<<<(unsigned)((tot4 + 255) / 256), 256, 0, stream>>>(ns, nv, out);

    const int nodeTiles = (NN + 15) / 16;      // 3125
    node_mlp<<<(nodeTiles + 7) / 8, 256, 0, stream>>>(ns, W1, b1, W2, b2, so);

    const long long edgeTiles = (EE + 15) / 16;  // 50000
    edge_msg<<<(unsigned)((edgeTiles + 7) / 8), 256, 0, stream>>>(nv, edge, ediff, edist,
                                                                  Wf, bf, so, outS, outV);
}